// PointLayerNorm_21294447854197
// MI455X (gfx1250) — compile-verified
//
#include <hip/hip_runtime.h>

// PointLayerNorm (segment layer-norm) for MI455X / gfx1250.
//
// Plan (HBM-bound: 307MB x read twice + 307MB out write ~= 0.92GB -> ~40us @ 23.3TB/s):
//   K0: zero 2*B*S accumulators in ws
//   K1: per-segment sum / sum-of-squares via V_WMMA_F32_16X16X4_F32 row-reduction
//       (A = 16 points x 4 channels, B = ones -> D rows = per-point partial sums),
//       accumulated over C/4 K-chunks, reduced via LDS f32 atomics, flushed with
//       sparse global f32 atomics (blocks own contiguous point ranges).
//   K2: 128-thread kernel: mean = S1/(cnt*C), inv = rsqrt(S2/(cnt*C) - mean^2 + eps)
//   K3: flat float4 apply: out = (x - mean)*inv*w + b
//
// Shapes: B=4, N=200000 (divisible by 16), C=96 (divisible by 4), S=32.

typedef __attribute__((ext_vector_type(2))) float v2f;
typedef __attribute__((ext_vector_type(8))) float v8f;

#if defined(__gfx1250__) && __has_builtin(__builtin_amdgcn_wmma_f32_16x16x4_f32)
#define USE_WMMA_F32X4 1
#else
#define USE_WMMA_F32X4 0
#endif

__global__ __launch_bounds__(256) void pln_zero(float* __restrict__ p, int n) {
    int i = blockIdx.x * blockDim.x + threadIdx.x;
    if (i < n) p[i] = 0.0f;
}

// Pass 1: segment moments. One wave handles 16 consecutive points per tile.
__global__ __launch_bounds__(256) void pln_moments(
    const float* __restrict__ x, const int* __restrict__ bidx,
    float* __restrict__ acc /* [2*BS]: sum | sumsq */,
    int N, int C, int S, int BS, int tiles, int tiles_per_wave)
{
    extern __shared__ float lmem[];               // 2*BS floats
    for (int i = threadIdx.x; i < 2 * BS; i += blockDim.x) lmem[i] = 0.0f;
    __syncthreads();

    const int lane = threadIdx.x & 31;            // wave32
    const int wg   = blockIdx.x * (blockDim.x >> 5) + (threadIdx.x >> 5);
    int t0 = wg * tiles_per_wave;
    int t1 = t0 + tiles_per_wave; if (t1 > tiles) t1 = tiles;

    const int pm = lane & 15;                     // point-in-tile owned by this lane
    const int KC = C >> 2;                        // K-chunks of 4 channels

    for (int t = t0; t < t1; ++t) {
        const long long p0 = (long long)t << 4;   // first global point of tile
        const int b  = (int)(p0 / N);
        const int n0 = (int)(p0 - (long long)b * N);   // tile start within batch
        const int seg = bidx[n0 + pm];                 // lane l holds seg of point l&15
        const float* base = x + ((long long)b * N + n0 + pm) * C;

#if USE_WMMA_F32X4
        // A-matrix 16x4 f32 layout: lanes 0-15 hold {K0,K1}, lanes 16-31 hold {K2,K3}.
        const float* lptr = base + ((lane >> 4) << 1);
        v8f cs = {0.f,0.f,0.f,0.f,0.f,0.f,0.f,0.f};
        v8f cq = {0.f,0.f,0.f,0.f,0.f,0.f,0.f,0.f};
        const v2f ones = {1.0f, 1.0f};
        for (int kk = 0; kk < KC; ++kk) {
            v2f a = *(const v2f*)(lptr + (kk << 2));
            cs = __builtin_amdgcn_wmma_f32_16x16x4_f32(false, a,      false, ones,
                                                       (short)0, cs, false, false);
            v2f a2 = a * a;
            cq = __builtin_amdgcn_wmma_f32_16x16x4_f32(false, a2,     false, ones,
                                                       (short)0, cq, false, false);
        }
        if (t + 1 < tiles)                        // next tile is +16*C floats (x contiguous)
            __builtin_prefetch(lptr + ((size_t)C << 4), 0, 3);

        // D layout: VGPR r = row r (lanes 0-15) / row r+8 (lanes 16-31); all cols equal.
        // Lane 0 extracts rows 0-7, lane 16 rows 8-15. Gather seg ids before diverging.
        const int hi = (lane >> 4) << 3;
        int segs[8];
        #pragma unroll
        for (int r = 0; r < 8; ++r) segs[r] = __shfl(seg, r + hi, 32);
        if ((lane & 15) == 0) {
            #pragma unroll
            for (int r = 0; r < 8; ++r) {
                const int g = b * S + segs[r];
                atomicAdd(&lmem[g],      cs[r]);
                atomicAdd(&lmem[BS + g], cq[r]);
            }
        }
#else
        // Fallback: two lanes per point (one per half-wave, C/2 channels each), shuffle combine.
        const int  CH = C >> 1;
        const float* pp = base + (lane >> 4) * CH;
        float s = 0.0f, q = 0.0f;
        for (int c = 0; c < CH; ++c) { float v = pp[c]; s += v; q += v * v; }
        s += __shfl_xor(s, 16, 32);
        q += __shfl_xor(q, 16, 32);
        if (lane < 16) {
            const int g = b * S + seg;
            atomicAdd(&lmem[g],      s);
            atomicAdd(&lmem[BS + g], q);
        }
#endif
    }

    __syncthreads();
    for (int i = threadIdx.x; i < 2 * BS; i += blockDim.x) {
        float v = lmem[i];
        if (v != 0.0f) atomicAdd(&acc[i], v);     // sparse: block spans ~1-3 segments
    }
}

// Pass 2: per-(b,s) mean + inv_std.
__global__ __launch_bounds__(128) void pln_stats(
    const float* __restrict__ acc, const int* __restrict__ offs,
    float* __restrict__ meanp, float* __restrict__ invp,
    int S, int BS, int C, float eps)
{
    int i = blockIdx.x * blockDim.x + threadIdx.x;
    if (i >= BS) return;
    int s = i % S;
    float cnt   = (float)(offs[s + 1] - offs[s]);
    float denom = cnt * (float)C;
    float m   = acc[i] / denom;
    float var = acc[BS + i] / denom - m * m;
    meanp[i] = m;
    invp[i]  = rsqrtf(var + eps);
}

// Pass 3: out = (x - mean) * inv * w + b, flat float4 stream.
__global__ __launch_bounds__(256) void pln_apply(
    const float4* __restrict__ x4, const int* __restrict__ bidx,
    const float4* __restrict__ w4, const float4* __restrict__ b4,
    const float* __restrict__ meanp, const float* __restrict__ invp,
    float4* __restrict__ out4, int N, int C4, int S, int total4)
{
    int f = blockIdx.x * blockDim.x + threadIdx.x;
    if (f >= total4) return;
    int p  = f / C4;                 // global point
    int cc = f - p * C4;             // float4 index within channels
    int b  = p / N;
    int n  = p - b * N;
    int g  = b * S + bidx[n];
    float m  = meanp[g];
    float iv = invp[g];
    float4 xv = x4[f], wv = w4[cc], bv = b4[cc], o;
    o.x = (xv.x - m) * iv * wv.x + bv.x;
    o.y = (xv.y - m) * iv * wv.y + bv.y;
    o.z = (xv.z - m) * iv * wv.z + bv.z;
    o.w = (xv.w - m) * iv * wv.w + bv.w;
    out4[f] = o;
}

extern "C" void kernel_launch(void* const* d_in, const int* in_sizes, int n_in,
                              void* d_out, int out_size, void* d_ws, size_t ws_size,
                              hipStream_t stream) {
    const float* x    = (const float*)d_in[0];
    const int*   offs = (const int*)d_in[1];   // JAX default int32
    const int*   bidx = (const int*)d_in[2];
    const float* w    = (const float*)d_in[3];
    const float* bias = (const float*)d_in[4];

    const int S = in_sizes[1] - 1;
    const int N = in_sizes[2];
    const int C = in_sizes[3];
    const long long total = (long long)in_sizes[0];        // B*N*C
    const int B  = (int)(total / ((long long)N * C));
    const int BS = B * S;

    float* ws    = (float*)d_ws;
    float* acc   = ws;             // [2*BS] sums
    float* meanp = ws + 2 * BS;    // [BS]
    float* invp  = ws + 3 * BS;    // [BS]

    // K0: zero accumulators (required every call; harness replays the graph).
    pln_zero<<<(2 * BS + 255) / 256, 256, 0, stream>>>(acc, 2 * BS);

    // K1: moments. Contiguous tile ranges per wave keep LDS->global flush sparse.
    const int tiles   = (int)(((long long)B * N) >> 4);    // N % 16 == 0 here
    const int blocks1 = 1024, thr1 = 256;
    const int waves   = blocks1 * (thr1 >> 5);
    const int tpw     = (tiles + waves - 1) / waves;
    const size_t lds  = (size_t)(2 * BS) * sizeof(float);
    pln_moments<<<blocks1, thr1, lds, stream>>>(x, bidx, acc, N, C, S, BS, tiles, tpw);

    // K2: stats (tiny).
    pln_stats<<<(BS + 127) / 128, 128, 0, stream>>>(acc, offs, meanp, invp, S, BS, C, 1e-5f);

    // K3: normalize + affine.
    const int C4     = C >> 2;                              // C % 4 == 0 here
    const int total4 = (int)(total >> 2);
    pln_apply<<<(total4 + 255) / 256, 256, 0, stream>>>(
        (const float4*)x, bidx, (const float4*)w, (const float4*)bias,
        meanp, invp, (float4*)d_out, N, C4, S, total4);
}